// HMM_74680891342934
// MI455X (gfx1250) — compile-verified
//
#include <hip/hip_runtime.h>
#include <hip/hip_bf16.h>

// ---------------- problem constants ----------------
static constexpr int Bn = 8, Tn = 400, Nn = 200, Dn = 80;
#define BIG_NEG  (-1e30f)
#define LOG_EPS  (-69.0775527898f)   // log(1e-30)
#define LN2      (0.69314718056f)
#define INV_LN2  (1.44269504089f)
// -0.5 * D * log(2*pi)
#define EMIT_CONST (-73.5150826466f)

typedef float v2f __attribute__((ext_vector_type(2)));
typedef float v4f __attribute__((ext_vector_type(4)));
typedef float v8f __attribute__((ext_vector_type(8)));

#define DEVFN static __device__ __forceinline__

// ---------------- wave32 cross-lane helpers (ds_bpermute, no LDS/barriers) --
DEVFN float lane_read_f(float v, int srcLane) {
    int r = __builtin_amdgcn_ds_bpermute(srcLane << 2, __float_as_int(v));
    return __int_as_float(r);
}
DEVFN float wave_max_f(float v) {
    int lane = (int)(threadIdx.x & 31u);
#pragma unroll
    for (int m = 16; m >= 1; m >>= 1)
        v = fmaxf(v, lane_read_f(v, lane ^ m));
    return v;
}
DEVFN float wave_sum_f(float v) {
    int lane = (int)(threadIdx.x & 31u);
#pragma unroll
    for (int m = 16; m >= 1; m >>= 1)
        v = v + lane_read_f(v, lane ^ m);
    return v;
}
DEVFN float lane_shift_up1(float v) {  // value from lane-1 (lane 0 -> wrapped, caller masks)
    int lane = (int)(threadIdx.x & 31u);
    return lane_read_f(v, (lane - 1) & 31);
}

// ---------------- fast math on native v_exp/v_log/v_rcp ----------------
DEVFN float fast_softplus(float x) {  // log(1+exp(x))
    float e = __builtin_amdgcn_exp2f(x * INV_LN2);
    float r = __builtin_amdgcn_logf(1.0f + e) * LN2;
    return (x > 15.0f) ? x : r;
}
DEVFN float fast_logaddexp(float a, float b) {
    float m = fmaxf(a, b);
    float d = fminf(a, b) - m;                       // <= 0 (0 if both BIG_NEG)
    float t = __builtin_amdgcn_logf(1.0f + __builtin_amdgcn_exp2f(d * INV_LN2)) * LN2;
    return m + t;
}

// ============================================================================
// Kernel 1: emission  logB[b,t,n] = state_mask * sum_d N(x_d | m, s) logprob
//   - 1 wave per (b,t); NT streaming loads of means/stds (410 MB, single use,
//     > 192 MB L2 -> keep them out of the way of the L2-resident logB)
//   - per-lane packed-f32 register accumulation in V_WMMA A-matrix layout
//   - one V_WMMA_F32_16X16X4_F32 with B=ones performs the complete horizontal
//     reduction (2 regs x 2 half-waves) for 16 states at once
//   - results written as 2x global_store_b128 from C[0..7] (branchless mask)
// ============================================================================
__global__ __launch_bounds__(256)
void hmm_emission_kernel(const float* __restrict__ mels,
                         const float* __restrict__ means,
                         const float* __restrict__ stds,
                         const int*   __restrict__ inputs_len,
                         float*       __restrict__ logB)
{
    const int wave = (int)(threadIdx.x >> 5);
    const int lane = (int)(threadIdx.x & 31u);
    const int bt   = (int)blockIdx.x * 8 + wave;          // exact: 400 blocks * 8 waves
    const int b    = bt / Tn;
    const int h    = lane >> 4;                           // half-wave: K = {2h, 2h+1}
    const int m16  = lane & 15;                           // state-in-tile (A-matrix M)
    const int len  = inputs_len[b];

    // Preload this lane's mel slice: d = 4*c + 2*h (+0,+1), c = 0..19
    const float* melp = mels + (size_t)bt * Dn;
    float xv[40];
#pragma unroll
    for (int c = 0; c < 20; ++c) {
        int d = 4 * c + 2 * h;
        xv[2 * c]     = melp[d];
        xv[2 * c + 1] = melp[d + 1];
    }

    v2f Bones; Bones.x = 1.0f; Bones.y = 1.0f;

#pragma unroll 1
    for (int tile = 0; tile < 13; ++tile) {               // 13*16 = 208 >= 200 states
        const int n0 = tile * 16;
        const int n  = n0 + m16;
        const int nc = (n < Nn) ? n : (Nn - 1);           // clamp so EXEC stays all-ones
        const float* mp = means + ((size_t)bt * Nn + nc) * Dn + 2 * h;
        const float* sp = stds  + ((size_t)bt * Nn + nc) * Dn + 2 * h;

        float a0 = 0.0f, a1 = 0.0f;                       // A-matrix elems K=2h, K=2h+1
#pragma unroll
        for (int c = 0; c < 20; ++c) {
            v2f m2 = __builtin_nontemporal_load((const v2f*)(mp + 4 * c));
            v2f s2 = __builtin_nontemporal_load((const v2f*)(sp + 4 * c));
            float i0 = __builtin_amdgcn_rcpf(s2.x);
            float i1 = __builtin_amdgcn_rcpf(s2.y);
            float z0 = (xv[2 * c]     - m2.x) * i0;
            float z1 = (xv[2 * c + 1] - m2.y) * i1;
            a0 = fmaf(z0 * z0, -0.5f, a0);
            a1 = fmaf(z1 * z1, -0.5f, a1);
            a0 = fmaf(__builtin_amdgcn_logf(s2.x), -LN2, a0);   // -= ln(std)
            a1 = fmaf(__builtin_amdgcn_logf(s2.y), -LN2, a1);
        }

        v2f A; A.x = a0; A.y = a1;
        v8f C = {};
        // out[i][j] = sum_k A[i][k]  (B = ones): full D-reduction for 16 states
        C = __builtin_amdgcn_wmma_f32_16x16x4_f32(false, A, false, Bones,
                                                  (short)0, C, false, false);

        // C layout: VGPR r, lanes0-15 -> M=r (N=lane), lanes16-31 -> M=r+8.
        // Lane 0 owns states n0..n0+7, lane 16 owns n0+8..n0+15.
        const int nsbase = n0 + h * 8;
        if (m16 == 0 && nsbase < Nn) {                    // false only for tile12/h==1
            v4f lo, hi;                                   // branchless state-mask select
#pragma unroll
            for (int r = 0; r < 4; ++r)
                lo[r] = ((nsbase + r)     < len) ? (C[r]     + EMIT_CONST) : 0.0f;
#pragma unroll
            for (int r = 0; r < 4; ++r)
                hi[r] = ((nsbase + 4 + r) < len) ? (C[4 + r] + EMIT_CONST) : 0.0f;
            v4f* dst = (v4f*)(logB + (size_t)bt * Nn + nsbase);   // 32B aligned
            dst[0] = lo;
            dst[1] = hi;
        }
    }
}

// ============================================================================
// Kernel 2: forward scan. One wave32 per batch, 7 states/lane, no barriers.
//  - renormalize by the running wave-max only (per-step c_t telescopes, final
//    answer identical), and only every 8 steps (range control for fp32)
//  - e/tv loads software-pipelined one step ahead (hide L2 latency)
//  - boundary shuffle (u[6] -> lane+1) issued at end of step t, consumed at
//    step t+1: ds_bpermute latency hidden under independent transcendentals
// ============================================================================
__global__ __launch_bounds__(32)
void hmm_scan_kernel(const float* __restrict__ logB,
                     const float* __restrict__ tvec,
                     const int*   __restrict__ inputs_len,
                     const int*   __restrict__ mel_lens,
                     float*       __restrict__ out)
{
    const int b    = (int)blockIdx.x;
    const int lane = (int)(threadIdx.x & 31u);
    const int len  = inputs_len[b];
    const int L    = mel_lens[b];                 // in [T/2, T], so L-1 >= 199 >= 1
    const float* eB  = logB + (size_t)b * Tn * Nn;
    const float* tvb = tvec + (size_t)b * Tn * Nn;

    constexpr int S = 7;                          // 7*32 = 224 >= N
    const int nbase = lane * S;
    const int npm1  = (nbase - 1 < 0) ? 0 : nbase - 1;   // predecessor state index

    float u[S];                                   // (lazily normalized) log-alpha
    float A;                                      // accumulated applied shifts

    // -------- t = 0 : priors + emission, normalize by max --------
    {
        float lmax = BIG_NEG;
#pragma unroll
        for (int i = 0; i < S; ++i) {
            int n = nbase + i;
            float e = (n < Nn) ? eB[n] : 0.0f;
            float v = (n < Nn) ? ((n == 0) ? e : (BIG_NEG + e)) : BIG_NEG;
            u[i] = v;
            lmax = fmaxf(lmax, v);
        }
        float M = wave_max_f(lmax);
        A = M;
#pragma unroll
        for (int i = 0; i < S; ++i) u[i] -= M;
    }
    float u6prev = lane_shift_up1(u[S - 1]);      // lane-1's last state, post-norm

    // prefetch step t=1
    float ee[S], tvs[S], tvp;
    {
        const float* e_t  = eB  + (size_t)Nn;     // t = 1
        const float* tv_t = tvb + (size_t)Nn;
#pragma unroll
        for (int i = 0; i < S; ++i) {
            int n = nbase + i;
            int nc = (n < Nn) ? n : (Nn - 1);
            ee[i]  = e_t[nc];
            tvs[i] = tv_t[nc];
        }
        tvp = tv_t[npm1];
    }

#pragma unroll 1
    for (int t = 1; t < Tn; ++t) {
        // ---- issue loads for t+1 (clamped; wasted only on the break step) ----
        float een[S], tvn[S], tvpn;
        {
            int tn = (t + 1 < Tn) ? (t + 1) : (Tn - 1);
            const float* e_t  = eB  + (size_t)tn * Nn;
            const float* tv_t = tvb + (size_t)tn * Nn;
#pragma unroll
            for (int i = 0; i < S; ++i) {
                int n = nbase + i;
                int nc = (n < Nn) ? n : (Nn - 1);
                een[i] = e_t[nc];
                tvn[i] = tv_t[nc];
            }
            tvpn = tv_t[npm1];
        }

        // ---- transitions for step t ----
        float stay[S], mv[S];
#pragma unroll
        for (int i = 0; i < S; ++i) {
            float tvv   = tvs[i];
            float sp    = fast_softplus(tvv);
            float lstay = fmaxf(-sp, LOG_EPS);          // log(sigmoid(-tv)) clamped
            float lmove = fmaxf(tvv - sp, LOG_EPS);     // log(sigmoid(+tv)) clamped
            stay[i] = u[i] + lstay;
            mv[i]   = u[i] + lmove;
        }
        // boundary "leaving" term from lane-1 (u6prev shuffled last iteration)
        float spp    = fast_softplus(tvp);
        float lmovep = fmaxf(tvp - spp, LOG_EPS);
        float leave0 = (lane == 0) ? BIG_NEG : (u6prev + lmovep);

        float lmax = BIG_NEG;
        float unew[S];
#pragma unroll
        for (int i = 0; i < S; ++i) {
            float leave = (i == 0) ? leave0 : mv[i - 1];
            float o = fast_logaddexp(stay[i], leave);
            int n = nbase + i;
            float v = (n < len) ? (o + ee[i]) : BIG_NEG;
            unew[i] = v;
            lmax = fmaxf(lmax, v);
        }

        // ---- lazy renormalization (every 8 steps; uniform branch) ----
        if ((t & 7) == 0) {
            float M = wave_max_f(lmax);
            A += M;
#pragma unroll
            for (int i = 0; i < S; ++i) u[i] = unew[i] - M;
        } else {
#pragma unroll
            for (int i = 0; i < S; ++i) u[i] = unew[i];
        }

        if (t == L - 1) {                               // uniform across the wave
            // absorption: logsumexp_n( la_last + [n==len-1]*log_move(tv_last) )
            float vmax = BIG_NEG;
            float val[S];
#pragma unroll
            for (int i = 0; i < S; ++i) {
                int n = nbase + i;
                float lmove = fmaxf(tvs[i] - fast_softplus(tvs[i]), LOG_EPS);
                float v = (n == len - 1) ? (u[i] + lmove) : BIG_NEG;
                val[i] = v;
                vmax = fmaxf(vmax, v);
            }
            float M2 = wave_max_f(vmax);
            float s = 0.0f;
#pragma unroll
            for (int i = 0; i < S; ++i)
                s += __builtin_amdgcn_exp2f((val[i] - M2) * INV_LN2);
            float Ssum = wave_sum_f(s);
            if (lane == 0)
                out[b] = A + M2 + __builtin_amdgcn_logf(Ssum) * LN2;
            return;
        }

        // shuffle issued here, consumed next iteration (latency hidden)
        u6prev = lane_shift_up1(u[S - 1]);

        // rotate prefetch buffers
#pragma unroll
        for (int i = 0; i < S; ++i) { ee[i] = een[i]; tvs[i] = tvn[i]; }
        tvp = tvpn;
    }
}

// ============================================================================
extern "C" void kernel_launch(void* const* d_in, const int* in_sizes, int n_in,
                              void* d_out, int out_size, void* d_ws, size_t ws_size,
                              hipStream_t stream) {
    const float* mels       = (const float*)d_in[0];
    const float* means      = (const float*)d_in[1];
    const float* stds       = (const float*)d_in[2];
    const float* tvec       = (const float*)d_in[3];
    const int*   inputs_len = (const int*)d_in[4];
    const int*   mel_lens   = (const int*)d_in[5];
    float*       out        = (float*)d_out;
    float*       logB       = (float*)d_ws;     // B*T*N floats = 2.56 MB (L2-resident)

    hmm_emission_kernel<<<(Bn * Tn) / 8, 256, 0, stream>>>(mels, means, stds,
                                                           inputs_len, logB);
    hmm_scan_kernel<<<Bn, 32, 0, stream>>>(logB, tvec, inputs_len, mel_lens, out);
}